// GCNEncoder_VAE_14929306321515
// MI455X (gfx1250) — compile-verified
//
#include <hip/hip_runtime.h>
#include <hip/hip_bf16.h>
#include <math.h>

// ---------------------------------------------------------------------------
// GENConv (softmax aggr) VGAE encoder for MI455X (gfx1250).
//  * Edge phase: 2 passes, L2-resident gathers + atomics (node table 51MB < 192MB L2)
//  * MLP phase: fp32 WMMA (v_wmma_f32_16x16x4_f32), weights staged in LDS in a
//    pair-interleaved layout so every B fragment is one aligned ds_load_b64,
//    BN stats fused via ds_add_f32 LDS atomics.
// ---------------------------------------------------------------------------

typedef float v2f __attribute__((ext_vector_type(2)));
typedef float v8f __attribute__((ext_vector_type(8)));

#define D_IN   128
#define D_HID  256
#define TILE_M 64

#define EPS_MSG 1e-7f
#define EPS_SM  1e-16f
#define EPS_BN  1e-5f

// ---------------------------------------------------------------------------
// Pass 1: per-(dst,feature) segment max of logits = t*(relu(x_src)+eps).
// logits > 0 (t=1), so uint atomicMax on the bit pattern == float max, and the
// memset-0 init matches the reference's where(isfinite(m), m, 0) for empty dsts.
// One wave per edge, each lane owns a float4 of the 128-wide feature row.
// ---------------------------------------------------------------------------
__global__ __launch_bounds__(256)
void edge_max_kernel(const float* __restrict__ h, const int* __restrict__ ei,
                     const float* __restrict__ t_all, int layer,
                     unsigned int* __restrict__ mbits, int E)
{
    int g = blockIdx.x * blockDim.x + threadIdx.x;
    int e = g >> 5, l = g & 31;
    if (e >= E) return;
    float t = t_all[layer];
    int src = ei[e];
    int dst = ei[E + e];
    const float4 xv = *(const float4*)(h + (size_t)src * D_IN + l * 4);
    float l0 = (fmaxf(xv.x, 0.f) + EPS_MSG) * t;
    float l1 = (fmaxf(xv.y, 0.f) + EPS_MSG) * t;
    float l2 = (fmaxf(xv.z, 0.f) + EPS_MSG) * t;
    float l3 = (fmaxf(xv.w, 0.f) + EPS_MSG) * t;
    unsigned int* mp = mbits + (size_t)dst * D_IN + l * 4;
    atomicMax(mp + 0, __float_as_uint(l0));
    atomicMax(mp + 1, __float_as_uint(l1));
    atomicMax(mp + 2, __float_as_uint(l2));
    atomicMax(mp + 3, __float_as_uint(l3));
}

// ---------------------------------------------------------------------------
// Pass 2: w = exp(logit - m[dst]); num[dst] += w*msg; den[dst] += w.
// ---------------------------------------------------------------------------
__global__ __launch_bounds__(256)
void edge_sum_kernel(const float* __restrict__ h, const int* __restrict__ ei,
                     const float* __restrict__ t_all, int layer,
                     const float* __restrict__ m,
                     float* __restrict__ num, float* __restrict__ den, int E)
{
    int g = blockIdx.x * blockDim.x + threadIdx.x;
    int e = g >> 5, l = g & 31;
    if (e >= E) return;
    float t = t_all[layer];
    int src = ei[e];
    int dst = ei[E + e];
    const float4 xv = *(const float4*)(h + (size_t)src * D_IN + l * 4);
    float ms0 = fmaxf(xv.x, 0.f) + EPS_MSG;
    float ms1 = fmaxf(xv.y, 0.f) + EPS_MSG;
    float ms2 = fmaxf(xv.z, 0.f) + EPS_MSG;
    float ms3 = fmaxf(xv.w, 0.f) + EPS_MSG;
    const float4 mm = *(const float4*)(m + (size_t)dst * D_IN + l * 4);
    float w0 = expf(ms0 * t - mm.x);
    float w1 = expf(ms1 * t - mm.y);
    float w2 = expf(ms2 * t - mm.z);
    float w3 = expf(ms3 * t - mm.w);
    size_t base = (size_t)dst * D_IN + l * 4;
    atomicAdd(&num[base + 0], w0 * ms0);
    atomicAdd(&num[base + 1], w1 * ms1);
    atomicAdd(&num[base + 2], w2 * ms2);
    atomicAdd(&num[base + 3], w3 * ms3);
    atomicAdd(&den[base + 0], w0);
    atomicAdd(&den[base + 1], w1);
    atomicAdd(&den[base + 2], w2);
    atomicAdd(&den[base + 3], w3);
}

// ---------------------------------------------------------------------------
// GEMM1 fused: hin = num/(den+eps) + h ; h1 = hin @ W1 (128->256); accumulate
// per-column sum / sumsq for BatchNorm. Persistent blocks: W1 staged once in
// LDS, pair-interleaved Wp[k/2][col][k%2] so a B fragment (K rows 2p,2p+1 of
// one column) is a single aligned ds_load_b64. 8 waves: 4 row strips x 2 col
// halves, each wave holds 8 f32 16x16 WMMA accumulator tiles.
// ---------------------------------------------------------------------------
__global__ __launch_bounds__(256)
void gemm1_bn_kernel(const float* __restrict__ h,
                     const float* __restrict__ num, const float* __restrict__ den,
                     const float* __restrict__ W1,     // [128,256] row-major
                     float* __restrict__ h1,           // [N,256]
                     float* __restrict__ gsum, float* __restrict__ gsumsq,
                     int N)
{
    extern __shared__ float lds[];
    float* W1p  = lds;                        // 64 pairs * 256 cols * 2
    float* hin  = W1p + D_IN * D_HID;         // 64*128
    float* csum = hin + TILE_M * D_IN;        // 256
    float* csq  = csum + D_HID;               // 256

    const int tid = threadIdx.x;
    // pair-interleave W1 into LDS: Wp[(p*256+col)*2 + (k&1)], p = k>>1
    for (int i = tid; i < (D_IN / 2) * D_HID; i += 256) {
        int p = i >> 8, col = i & 255;
        v2f w;
        w.x = W1[(2 * p + 0) * D_HID + col];
        w.y = W1[(2 * p + 1) * D_HID + col];
        *(v2f*)&W1p[(size_t)i * 2] = w;
    }
    csum[tid] = 0.f; csq[tid] = 0.f;
    __syncthreads();

    const int wave = tid >> 5, lane = tid & 31;
    const int rs   = (wave & 3) * 16;   // row strip within tile
    const int ch   = (wave >> 2) * 128; // column half
    const int lrow = lane & 15;
    const int lhi  = lane >> 4;

    const int ntiles = (N + TILE_M - 1) / TILE_M;
    for (int tile = blockIdx.x; tile < ntiles; tile += gridDim.x) {
        // stage hin = aggr + h (zero-pad OOB rows; zero rows add 0 to BN sums)
        for (int j = tid; j < TILE_M * D_IN; j += 256) {
            int r = j >> 7, c = j & 127;
            int gr = tile * TILE_M + r;
            float v = 0.f;
            if (gr < N) {
                size_t gi = (size_t)gr * D_IN + c;
                v = num[gi] / (den[gi] + EPS_SM) + h[gi];
            }
            hin[j] = v;
        }
        __syncthreads();

        v8f acc[8] = {};
        for (int k0 = 0; k0 < D_IN; k0 += 4) {
            // A frag (16x4 f32): lane l -> row l&15, K = k0 + 2*(l>>4) + {0,1}
            v2f a = *(const v2f*)&hin[(rs + lrow) * D_IN + k0 + (lhi << 1)];
            // B frags first (batch the ds loads), then the WMMA chain
            const int p = (k0 >> 1) + lhi;   // pair index = (k0 + 2*lhi)/2
            v2f bf[8];
            #pragma unroll
            for (int ct = 0; ct < 8; ++ct)
                bf[ct] = *(const v2f*)&W1p[((size_t)p * D_HID + (ch + ct * 16 + lrow)) * 2];
            #pragma unroll
            for (int ct = 0; ct < 8; ++ct)
                acc[ct] = __builtin_amdgcn_wmma_f32_16x16x4_f32(
                    false, a, false, bf[ct], (short)0, acc[ct], false, false);
        }

        // store h1 + per-column BN partial sums (C/D: VGPR v -> M = v + 8*(l>>4))
        #pragma unroll
        for (int ct = 0; ct < 8; ++ct) {
            int colg = ch + ct * 16 + lrow;
            float s = 0.f, sq = 0.f;
            #pragma unroll
            for (int v = 0; v < 8; ++v) {
                float val = acc[ct][v];
                int gr = tile * TILE_M + rs + v + (lhi << 3);
                if (gr < N) h1[(size_t)gr * D_HID + colg] = val;
                s += val; sq += val * val;
            }
            atomicAdd(&csum[colg], s);   // ds_add_f32
            atomicAdd(&csq[colg], sq);
        }
        __syncthreads();
    }
    atomicAdd(&gsum[tid], csum[tid]);
    atomicAdd(&gsumsq[tid], csq[tid]);
}

// ---------------------------------------------------------------------------
// BN finalize: scale/shift per hidden column.
// ---------------------------------------------------------------------------
__global__ void bn_finalize_kernel(const float* __restrict__ gsum,
                                   const float* __restrict__ gsumsq,
                                   const float* __restrict__ gamma,
                                   const float* __restrict__ beta,
                                   float* __restrict__ scaleshift, int N)
{
    int c = threadIdx.x;  // 256
    float invN = 1.f / (float)N;
    float mean = gsum[c] * invN;
    float var  = fmaxf(gsumsq[c] * invN - mean * mean, 0.f);
    float sc   = gamma[c] * rsqrtf(var + EPS_BN);
    scaleshift[c]          = sc;
    scaleshift[D_HID + c]  = beta[c] - mean * sc;
}

// ---------------------------------------------------------------------------
// GEMM2 fused: out = [relu_out]( relu(h1*scale+shift) @ W2 ) (256->128).
// W2 pair-interleaved in LDS. 8 waves: 4 row strips x 2 col halves of 64,
// 4 accum tiles each, K=256.
// ---------------------------------------------------------------------------
__global__ __launch_bounds__(256)
void gemm2_kernel(const float* __restrict__ h1,
                  const float* __restrict__ W2,          // [256,128] row-major
                  const float* __restrict__ scaleshift,  // scale[256], shift[256]
                  float* __restrict__ out, int N, int relu_out)
{
    extern __shared__ float lds[];
    float* W2p = lds;                         // 128 pairs * 128 cols * 2
    float* s1  = W2p + D_HID * D_IN;          // 64*256
    float* scl = s1 + TILE_M * D_HID;         // 256
    float* sht = scl + D_HID;                 // 256

    const int tid = threadIdx.x;
    for (int i = tid; i < (D_HID / 2) * D_IN; i += 256) {
        int p = i >> 7, col = i & 127;
        v2f w;
        w.x = W2[(2 * p + 0) * D_IN + col];
        w.y = W2[(2 * p + 1) * D_IN + col];
        *(v2f*)&W2p[(size_t)i * 2] = w;
    }
    scl[tid] = scaleshift[tid];
    sht[tid] = scaleshift[D_HID + tid];
    __syncthreads();

    const int wave = tid >> 5, lane = tid & 31;
    const int rs   = (wave & 3) * 16;
    const int cq   = (wave >> 2) * 64;
    const int lrow = lane & 15;
    const int lhi  = lane >> 4;

    const int ntiles = (N + TILE_M - 1) / TILE_M;
    for (int tile = blockIdx.x; tile < ntiles; tile += gridDim.x) {
        for (int j = tid; j < TILE_M * D_HID; j += 256) {
            int r = j >> 8, c = j & 255;
            int gr = tile * TILE_M + r;
            float v = 0.f;
            if (gr < N)
                v = fmaxf(h1[(size_t)gr * D_HID + c] * scl[c] + sht[c], 0.f);
            s1[j] = v;
        }
        __syncthreads();

        v8f acc[4] = {};
        for (int k0 = 0; k0 < D_HID; k0 += 4) {
            v2f a = *(const v2f*)&s1[(rs + lrow) * D_HID + k0 + (lhi << 1)];
            const int p = (k0 >> 1) + lhi;
            v2f bf[4];
            #pragma unroll
            for (int ct = 0; ct < 4; ++ct)
                bf[ct] = *(const v2f*)&W2p[((size_t)p * D_IN + (cq + ct * 16 + lrow)) * 2];
            #pragma unroll
            for (int ct = 0; ct < 4; ++ct)
                acc[ct] = __builtin_amdgcn_wmma_f32_16x16x4_f32(
                    false, a, false, bf[ct], (short)0, acc[ct], false, false);
        }

        #pragma unroll
        for (int ct = 0; ct < 4; ++ct) {
            int colg = cq + ct * 16 + lrow;
            #pragma unroll
            for (int v = 0; v < 8; ++v) {
                float val = acc[ct][v];
                if (relu_out) val = fmaxf(val, 0.f);
                int gr = tile * TILE_M + rs + v + (lhi << 3);
                if (gr < N) out[(size_t)gr * D_IN + colg] = val;
            }
        }
        __syncthreads();
    }
}

// ---------------------------------------------------------------------------
// Host-side orchestration (all on `stream`, graph-capture safe).
// ---------------------------------------------------------------------------
extern "C" void kernel_launch(void* const* d_in, const int* in_sizes, int n_in,
                              void* d_out, int out_size, void* d_ws, size_t ws_size,
                              hipStream_t stream) {
    const float* x         = (const float*)d_in[0];
    const int*   ei        = (const int*)d_in[1];   // [2,E]: src row then dst row
    const float* t_all     = (const float*)d_in[2];
    const float* W1_all    = (const float*)d_in[3]; // [4,128,256]
    const float* W2_all    = (const float*)d_in[4]; // [4,256,128]
    const float* gamma_all = (const float*)d_in[5]; // [4,256]
    const float* beta_all  = (const float*)d_in[6]; // [4,256]

    const int N = in_sizes[0] / D_IN;
    const int E = in_sizes[1] / 2;

    float* ws = (float*)d_ws;
    const size_t ND = (size_t)N * D_IN;
    float* w_m    = ws;                 // [N,128] logit max (uint bits)
    float* w_num  = ws + ND;            // [N,128]
    float* w_den  = ws + 2 * ND;        // [N,128]
    float* w_h1   = ws + 3 * ND;        // [N,256]
    float* w_hA   = ws + 5 * ND;        // [N,128]
    float* w_hB   = ws + 6 * ND;        // [N,128]
    float* w_stat = ws + 7 * ND;        // gsum[256], gsumsq[256], scale/shift[512]
    float* w_gsum   = w_stat;
    float* w_gsumsq = w_stat + D_HID;
    float* w_ss     = w_stat + 2 * D_HID;

    const int edge_blocks = (E * 32 + 255) / 256;
    const int gemm_blocks = 512;
    const size_t lds1 = (size_t)(D_IN * D_HID + TILE_M * D_IN + 2 * D_HID) * 4;   // ~166 KB
    const size_t lds2 = (size_t)(D_HID * D_IN + TILE_M * D_HID + 2 * D_HID) * 4;  // ~199 KB

    auto genconv = [&](const float* hin, int layer, float* hout, int relu_out) {
        hipMemsetAsync(w_m, 0, 3 * ND * sizeof(float), stream);        // m,num,den
        hipMemsetAsync(w_gsum, 0, 2 * D_HID * sizeof(float), stream);  // BN sums
        edge_max_kernel<<<edge_blocks, 256, 0, stream>>>(
            hin, ei, t_all, layer, (unsigned int*)w_m, E);
        edge_sum_kernel<<<edge_blocks, 256, 0, stream>>>(
            hin, ei, t_all, layer, w_m, w_num, w_den, E);
        gemm1_bn_kernel<<<gemm_blocks, 256, lds1, stream>>>(
            hin, w_num, w_den, W1_all + (size_t)layer * D_IN * D_HID,
            w_h1, w_gsum, w_gsumsq, N);
        bn_finalize_kernel<<<1, 256, 0, stream>>>(
            w_gsum, w_gsumsq, gamma_all + (size_t)layer * D_HID,
            beta_all + (size_t)layer * D_HID, w_ss, N);
        gemm2_kernel<<<gemm_blocks, 256, lds2, stream>>>(
            w_h1, W2_all + (size_t)layer * D_HID * D_IN, w_ss, hout, N, relu_out);
    };

    float* mu     = (float*)d_out;
    float* logstd = (float*)d_out + ND;

    genconv(x,    0, w_hA,   1);  // layer 0 + outer relu
    genconv(w_hA, 1, w_hB,   1);  // layer 1 + outer relu
    genconv(w_hB, 2, mu,     0);  // mu head
    genconv(w_hB, 3, logstd, 0);  // logstd head
}